// RHO_local_31645319037048
// MI455X (gfx1250) — compile-verified
//
#include <hip/hip_runtime.h>

#define N_NODES 50000
#define N_EDGES 600000
#define D 128
#define LAYERS 2

typedef __attribute__((ext_vector_type(2))) float v2f;
typedef __attribute__((ext_vector_type(8))) float v8f;

// ---------------------------------------------------------------------------
// Kernel A: H = Xin  (vectorized copy; H then accumulates the -temp*Lap term)
// ---------------------------------------------------------------------------
__global__ void rho_copy_kernel(const float* __restrict__ src,
                                float* __restrict__ dst, int n4) {
    int i = blockIdx.x * blockDim.x + threadIdx.x;
    if (i < n4) {
        ((float4*)dst)[i] = ((const float4*)src)[i];
    }
}

// ---------------------------------------------------------------------------
// Kernel B: fused scatter:  H[row[e]][d] += -temps[d] * vals[e] * X[col[e]][d]
// One wave32 per edge; lane L covers features L, L+32, L+64, L+96.
// Edge index is wave-uniform -> row/col/vals become scalar loads; the feature
// accesses are coalesced 128B full-wave bursts. X (25.6MB) lives in L2.
// ---------------------------------------------------------------------------
__global__ void rho_scatter_kernel(const int* __restrict__ row,
                                   const int* __restrict__ col,
                                   const float* __restrict__ vals,
                                   const float* __restrict__ X,
                                   const float* __restrict__ temp,
                                   float* __restrict__ H) {
    int idx  = blockIdx.x * blockDim.x + threadIdx.x;
    int lane = idx & 31;
    int e    = idx >> 5;              // wave-uniform
    if (e >= N_EDGES) return;
    int   r = row[e];
    int   c = col[e];
    float v = vals[e];
    const float* xr = X + (long)c * D;
    float*       hr = H + (long)r * D;
#pragma unroll
    for (int j = 0; j < 4; ++j) {
        int d = lane + 32 * j;
        float contrib = -temp[d] * v * xr[d];
        atomicAdd(&hr[d], contrib);
    }
}

// ---------------------------------------------------------------------------
// Kernel C: Y = relu(H @ W^T + bias) via V_WMMA_F32_16X16X4_F32 (exact fp32),
// fed from LDS staged with CDNA5 async global->LDS copies (ASYNCcnt path).
//
// Block = 256 thr = 8 waves; block owns one 16-row tile of H (8 KB) and the
// whole 128x128 W matrix (64 KB), both staged into LDS with
// GLOBAL_LOAD_ASYNC_TO_LDS_B128. The 24-bit IOFFSET is added to BOTH the LDS
// and global addresses, so one base pair + literal offsets covers each region.
// Wave w computes the 16x16 tile at columns [16w, 16w+16).
//
// A-layout (ISA 7.12.2, 32-bit A 16x4):  lane L, VGPR j holds
//   A[m = L&15][k0 + 2*(L>>4) + j]      -> contiguous float2 (ds_load_b64).
// B 4x16 mirrors it with N striped across lanes (GEMM is H @ W^T):
//   B[k][n = L&15] = W[nbase + (L&15)][k]  -> contiguous float2 per lane.
// C/D:  c[v] = out[m = v + 8*(L>>4)][n = L&15].
// ---------------------------------------------------------------------------
__global__ void rho_wmma_gemm_kernel(const float* __restrict__ H,
                                     const float* __restrict__ W,
                                     const float* __restrict__ bias,
                                     float* __restrict__ Y) {
    __shared__ float shH[16 * D];    //  8 KB : block's A tile
    __shared__ float shW[D * D];     // 64 KB : full weight matrix

    const int t     = threadIdx.x;   // 0..255
    const int lane  = t & 31;
    const int wave  = t >> 5;        // 0..7 -> column tile
    const int mbase = blockIdx.x * 16;   // 3125 row tiles (50000/16 exact)

    // ---- async stage H tile: 2048 floats, 256 lanes x 16B x 2 rounds ----
    {
        const float*  gsrc = H + (size_t)mbase * D + t * 4;
        unsigned int  ldst = (unsigned int)(size_t)(&shH[t * 4]);
        asm volatile(
            "global_load_async_to_lds_b128 %0, %1, off offset:0\n\t"
            "global_load_async_to_lds_b128 %0, %1, off offset:4096"
            :: "v"(ldst), "v"(gsrc) : "memory");
    }
    // ---- async stage W: 16384 floats, 256 lanes x 16B x 16 rounds ----
    {
        const float*  gsrc = W + t * 4;
        unsigned int  ldst = (unsigned int)(size_t)(&shW[t * 4]);
        asm volatile(
            "global_load_async_to_lds_b128 %0, %1, off offset:0\n\t"
            "global_load_async_to_lds_b128 %0, %1, off offset:4096\n\t"
            "global_load_async_to_lds_b128 %0, %1, off offset:8192\n\t"
            "global_load_async_to_lds_b128 %0, %1, off offset:12288\n\t"
            "global_load_async_to_lds_b128 %0, %1, off offset:16384\n\t"
            "global_load_async_to_lds_b128 %0, %1, off offset:20480\n\t"
            "global_load_async_to_lds_b128 %0, %1, off offset:24576\n\t"
            "global_load_async_to_lds_b128 %0, %1, off offset:28672\n\t"
            "global_load_async_to_lds_b128 %0, %1, off offset:32768\n\t"
            "global_load_async_to_lds_b128 %0, %1, off offset:36864\n\t"
            "global_load_async_to_lds_b128 %0, %1, off offset:40960\n\t"
            "global_load_async_to_lds_b128 %0, %1, off offset:45056\n\t"
            "global_load_async_to_lds_b128 %0, %1, off offset:49152\n\t"
            "global_load_async_to_lds_b128 %0, %1, off offset:53248\n\t"
            "global_load_async_to_lds_b128 %0, %1, off offset:57344\n\t"
            "global_load_async_to_lds_b128 %0, %1, off offset:61440"
            :: "v"(ldst), "v"(gsrc) : "memory");
    }
    // Each wave waits for its own async copies, then block-sync for LDS.
    asm volatile("s_wait_asynccnt 0x0" ::: "memory");
    __syncthreads();

    const int half = lane >> 4;          // 0 or 1
    const int l15  = lane & 15;
    const int koff = 2 * half;

    const float* arow = &shH[l15 * D];              // A: row m = lane&15
    const float* brow = &shW[(wave * 16 + l15) * D]; // B: W row n = lane&15

    v8f c = {};
#pragma unroll 8
    for (int k0 = 0; k0 < D; k0 += 4) {
        v2f a = *(const v2f*)(arow + k0 + koff);
        v2f b = *(const v2f*)(brow + k0 + koff);
        // 8 args: (neg_a, A, neg_b, B, c_mod, C, reuse_a, reuse_b)
        c = __builtin_amdgcn_wmma_f32_16x16x4_f32(
                /*neg_a=*/false, a, /*neg_b=*/false, b,
                /*c_mod=*/(short)0, c, /*reuse_a=*/false, /*reuse_b=*/false);
    }

    const float bv = bias[wave * 16 + l15];
    float* yb = Y + (size_t)mbase * D + wave * 16 + l15;
#pragma unroll
    for (int v = 0; v < 8; ++v) {
        int   m = v + 8 * half;
        float o = c[v] + bv;
        o = o > 0.0f ? o : 0.0f;           // ReLU
        yb[(size_t)m * D] = o;
    }
}

// ---------------------------------------------------------------------------
// Host launcher
// ---------------------------------------------------------------------------
extern "C" void kernel_launch(void* const* d_in, const int* in_sizes, int n_in,
                              void* d_out, int out_size, void* d_ws, size_t ws_size,
                              hipStream_t stream) {
    (void)in_sizes; (void)n_in; (void)out_size; (void)ws_size;

    const int*   row   = (const int*)  d_in[0];
    const int*   col   = (const int*)  d_in[1];
    const float* vals  = (const float*)d_in[2];
    const float* X0    = (const float*)d_in[3];
    const float* temps = (const float*)d_in[4];
    const float* Wt    = (const float*)d_in[5];
    const float* bb    = (const float*)d_in[6];

    float* out = (float*)d_out;                 // N_NODES * D
    float* H   = (float*)d_ws;                  // N_NODES * D scratch (25.6 MB)

    const int n4 = N_NODES * D / 4;
    dim3 cb(256), cg((n4 + 255) / 256);
    dim3 sb(256), sg((N_EDGES * 32 + 255) / 256);
    dim3 gb(256), gg(N_NODES / 16);

    for (int layer = 0; layer < LAYERS; ++layer) {
        const float* Xin = (layer == 0) ? X0 : out;
        rho_copy_kernel<<<cg, cb, 0, stream>>>(Xin, H, n4);
        rho_scatter_kernel<<<sg, sb, 0, stream>>>(row, col, vals, Xin,
                                                  temps + layer * D, H);
        rho_wmma_gemm_kernel<<<gg, gb, 0, stream>>>(H, Wt + (long)layer * D * D,
                                                    bb + layer * D, out);
    }
}